// OtherEmbedding_4776003633688
// MI455X (gfx1250) — compile-verified
//
#include <hip/hip_runtime.h>

typedef float v2f __attribute__((ext_vector_type(2)));
typedef float v8f __attribute__((ext_vector_type(8)));

#define EPS_LN 1e-5f

static __device__ __forceinline__ float leaky02(float x) {
  return x > 0.0f ? x : 0.2f * x;
}

static __device__ __forceinline__ void atomic_max_f32(float* addr, float val) {
  unsigned int* ua = reinterpret_cast<unsigned int*>(addr);
  unsigned int cur = __float_as_uint(*addr);
  while (__uint_as_float(cur) < val) {
    unsigned int prev = atomicCAS(ua, cur, __float_as_uint(val));
    if (prev == cur) break;
    cur = prev;
  }
}

// ---------------------------------------------------------------------------
// GEMM: C[M,256] = A[M,128] @ W[128,256]   (accumulate!=0 -> C += A@W)
// Block = 256 threads (8 waves, wave32). Block tile: 256 rows x 64 cols.
// Each wave: 32 rows x 64 cols = 2x4 accumulators of v_wmma_f32_16x16x4_f32,
// so every B fragment feeds two WMMAs (halves LDS->VGPR traffic per FLOP).
//
// LDS B layout is K-pair interleaved: ldsB[(k>>1)*128 + n*2 + (k&1)], so the
// WMMA B operand (W[kk][n], W[kk+1][n]) is one aligned ds_load_b64 directly
// into an even VGPR pair -- no repacking movs in the hot loop.
//
// WMMA f32 16x16x4 operand layouts (ISA 7.12.2):
//   A: lane(l&15) = M row; lanes 0-15 carry K=k0,k0+1; lanes 16-31 K=k0+2,k0+3
//   B: V0 = row K (lanes 0-15) / K+2 (lanes 16-31), V1 = K+1 / K+3
//   C/D: VGPR r holds rows M=r (lanes 0-15) and M=r+8 (lanes 16-31)
// ---------------------------------------------------------------------------
__global__ __launch_bounds__(256) void gemm128x256_wmma(
    const float* __restrict__ A, const float* __restrict__ W,
    float* __restrict__ C, int M, int accumulate)
{
  __shared__ float ldsB[128 * 64];   // 32 KB W tile, K-pair interleaved
  const int tid  = threadIdx.x;
  const int wave = tid >> 5;
  const int lane = tid & 31;
  const int lo   = lane & 15;
  const int hi   = lane >> 4;
  const int nblk = blockIdx.y;       // 4 column blocks of 64
  const int mblk = blockIdx.x;

  // Stage W tile: read two consecutive K rows (float4 each), write K-pair
  // interleaved as four aligned b64 stores.
  #pragma unroll
  for (int i = 0; i < 4; ++i) {
    int item = i * 256 + tid;                // 1024 items: 64 kpairs x 16 n-chunks
    int kp = item >> 4;                      // K pair index 0..63
    int nc = (item & 15) << 2;               // n chunk 0,4,...,60
    const float* wg = W + (size_t)(2 * kp) * 256 + nblk * 64 + nc;
    float4 w0 = *(const float4*)(wg);
    float4 w1 = *(const float4*)(wg + 256);
    float* bp = &ldsB[kp * 128 + nc * 2];
    v2f p0; p0.x = w0.x; p0.y = w1.x;
    v2f p1; p1.x = w0.y; p1.y = w1.y;
    v2f p2; p2.x = w0.z; p2.y = w1.z;
    v2f p3; p3.x = w0.w; p3.y = w1.w;
    *(v2f*)(bp + 0) = p0;
    *(v2f*)(bp + 2) = p1;
    *(v2f*)(bp + 4) = p2;
    *(v2f*)(bp + 6) = p3;
  }
  __syncthreads();

  const int rowbase = mblk * 256 + wave * 32;
  int r0 = rowbase + lo;
  int r1 = rowbase + 16 + lo;
  const float* A0 = A + (size_t)(r0 < M ? r0 : M - 1) * 128 + hi * 2;
  const float* A1 = A + (size_t)(r1 < M ? r1 : M - 1) * 128 + hi * 2;

  v8f acc0[4] = {};
  v8f acc1[4] = {};

  #pragma unroll 4
  for (int k0 = 0; k0 < 128; k0 += 4) {
    v2f a0 = *(const v2f*)(A0 + k0);         // K = k0+hi*2, +1
    v2f a1 = *(const v2f*)(A1 + k0);
    const int kp = (k0 >> 1) + hi;           // covers K = k0+2*hi, k0+2*hi+1
    const float* bb = &ldsB[kp * 128 + lo * 2];
    #pragma unroll
    for (int t = 0; t < 4; ++t) {
      v2f b = *(const v2f*)(bb + t * 32);    // n = t*16+lo
      acc0[t] = __builtin_amdgcn_wmma_f32_16x16x4_f32(
          false, a0, false, b, (short)0, acc0[t], false, false);
      acc1[t] = __builtin_amdgcn_wmma_f32_16x16x4_f32(
          false, a1, false, b, (short)0, acc1[t], false, false);
    }
  }

  #define STORE_STRIP(ACC, STRIPOFF)                                      \
    {                                                                     \
      const int baserow = rowbase + (STRIPOFF) + hi * 8;                  \
      _Pragma("unroll")                                                   \
      for (int r = 0; r < 8; ++r) {                                       \
        int rr = baserow + r;                                             \
        if (rr < M) {                                                     \
          float* crow = C + (size_t)rr * 256 + nblk * 64 + lo;            \
          _Pragma("unroll")                                               \
          for (int t = 0; t < 4; ++t) {                                   \
            float v = ACC[t][r];                                          \
            if (accumulate) v += crow[t * 16];                            \
            crow[t * 16] = v;                                             \
          }                                                               \
        }                                                                 \
      }                                                                   \
    }
  STORE_STRIP(acc0, 0)
  STORE_STRIP(acc1, 16)
  #undef STORE_STRIP
}

// ---------------------------------------------------------------------------
// Wa[k][h] = sum_c W[k][h*64+c] * a[h][c]  (collapses attention vector into
// the input space so al = x @ Wa; hd never needs materializing)
// ---------------------------------------------------------------------------
__global__ void compute_wa_kernel(const float* __restrict__ W,
                                  const float* __restrict__ a_src,
                                  const float* __restrict__ a_dst,
                                  float* __restrict__ Wa_s,
                                  float* __restrict__ Wa_d)
{
  int k = threadIdx.x;   // blockDim = 128
  #pragma unroll
  for (int h = 0; h < 4; ++h) {
    float s = 0.0f, d = 0.0f;
    for (int c = 0; c < 64; ++c) {
      float w = W[(size_t)k * 256 + h * 64 + c];
      s += w * a_src[h * 64 + c];
      d += w * a_dst[h * 64 + c];
    }
    Wa_s[k * 4 + h] = s;
    Wa_d[k * 4 + h] = d;
  }
}

// ---------------------------------------------------------------------------
// al[n][h] = dot(x[n,:128], Wa[:,h]); one wave per node, lane owns 4 K slots,
// shfl_xor tree reduction (wave32).
// ---------------------------------------------------------------------------
__global__ __launch_bounds__(256) void attn_logits_kernel(
    const float* __restrict__ X, const float* __restrict__ Wa,
    float* __restrict__ al, int N)
{
  const int wave = threadIdx.x >> 5;
  const int lane = threadIdx.x & 31;
  const int node = blockIdx.x * 8 + wave;
  if (node >= N) return;

  float4 xv = *(const float4*)(X + (size_t)node * 128 + lane * 4);
  const float xr[4] = { xv.x, xv.y, xv.z, xv.w };
  const float* wk = Wa + lane * 16;   // 4 K rows * 4 heads
  float s0 = 0.f, s1 = 0.f, s2 = 0.f, s3 = 0.f;
  #pragma unroll
  for (int j = 0; j < 4; ++j) {
    float xx = xr[j];
    s0 += xx * wk[j * 4 + 0];
    s1 += xx * wk[j * 4 + 1];
    s2 += xx * wk[j * 4 + 2];
    s3 += xx * wk[j * 4 + 3];
  }
  #pragma unroll
  for (int off = 16; off >= 1; off >>= 1) {
    s0 += __shfl_xor(s0, off, 32);
    s1 += __shfl_xor(s1, off, 32);
    s2 += __shfl_xor(s2, off, 32);
    s3 += __shfl_xor(s3, off, 32);
  }
  if (lane == 0) {
    float4 o = { s0, s1, s2, s3 };
    *(float4*)(al + (size_t)node * 4) = o;
  }
}

__global__ void init_mz_kernel(float* __restrict__ m, float* __restrict__ z, int n)
{
  int i = blockIdx.x * blockDim.x + threadIdx.x;
  if (i < n) { m[i] = -__builtin_inff(); z[i] = 0.0f; }
}

__global__ void init_bias_kernel(float* __restrict__ out,
                                 const float* __restrict__ b0,
                                 const float* __restrict__ b1, int n)
{
  int i = blockIdx.x * blockDim.x + threadIdx.x;
  if (i < n) {
    int c = i & 255;
    float v = b0[c];
    if (b1) v += b1[c];
    out[i] = v;
  }
}

// Pass 1: segment max of leaky-relu logits per (dst, head)
__global__ __launch_bounds__(256) void edge_max_kernel(
    const int* __restrict__ src, const int* __restrict__ dst,
    const float* __restrict__ al_s, const float* __restrict__ al_d,
    float* __restrict__ m, int E)
{
  int e = blockIdx.x * blockDim.x + threadIdx.x;
  if (e >= E) return;
  int s = src[e], d = dst[e];
  float4 a = *(const float4*)(al_s + (size_t)s * 4);
  float4 b = *(const float4*)(al_d + (size_t)d * 4);
  atomic_max_f32(&m[d * 4 + 0], leaky02(a.x + b.x));
  atomic_max_f32(&m[d * 4 + 1], leaky02(a.y + b.y));
  atomic_max_f32(&m[d * 4 + 2], leaky02(a.z + b.z));
  atomic_max_f32(&m[d * 4 + 3], leaky02(a.w + b.w));
}

// Pass 2: z[dst,h] += exp(logit - m[dst,h])
__global__ __launch_bounds__(256) void edge_expsum_kernel(
    const int* __restrict__ src, const int* __restrict__ dst,
    const float* __restrict__ al_s, const float* __restrict__ al_d,
    const float* __restrict__ m, float* __restrict__ z, int E)
{
  int e = blockIdx.x * blockDim.x + threadIdx.x;
  if (e >= E) return;
  int s = src[e], d = dst[e];
  float4 a = *(const float4*)(al_s + (size_t)s * 4);
  float4 b = *(const float4*)(al_d + (size_t)d * 4);
  float4 mm = *(const float4*)(m + (size_t)d * 4);
  atomicAdd(&z[d * 4 + 0], expf(leaky02(a.x + b.x) - mm.x));
  atomicAdd(&z[d * 4 + 1], expf(leaky02(a.y + b.y) - mm.y));
  atomicAdd(&z[d * 4 + 2], expf(leaky02(a.z + b.z) - mm.z));
  atomicAdd(&z[d * 4 + 3], expf(leaky02(a.w + b.w) - mm.w));
}

// Pass 3: out[dst,:] += alpha[e,h] * hs[src,:].  One wave per edge; each lane
// owns 8 consecutive channels (all inside one head), float4 gathers from hs.
__global__ __launch_bounds__(256) void edge_msg_kernel(
    const int* __restrict__ src, const int* __restrict__ dst,
    const float* __restrict__ al_s, const float* __restrict__ al_d,
    const float* __restrict__ m, const float* __restrict__ z,
    const float* __restrict__ hs, float* __restrict__ out, int total)
{
  int idx = blockIdx.x * blockDim.x + threadIdx.x;
  if (idx >= total) return;
  int e    = idx >> 5;
  int part = idx & 31;
  int c0   = part * 8;
  int h    = c0 >> 6;
  int s = src[e], d = dst[e];
  float logit = leaky02(al_s[(size_t)s * 4 + h] + al_d[(size_t)d * 4 + h]);
  float alpha = expf(logit - m[(size_t)d * 4 + h]) / (z[(size_t)d * 4 + h] + 1e-16f);
  const float4* hp = (const float4*)(hs + (size_t)s * 256 + c0);
  float4 h0 = hp[0], h1 = hp[1];
  float* op = out + (size_t)d * 256 + c0;
  atomicAdd(op + 0, alpha * h0.x);
  atomicAdd(op + 1, alpha * h0.y);
  atomicAdd(op + 2, alpha * h0.z);
  atomicAdd(op + 3, alpha * h0.w);
  atomicAdd(op + 4, alpha * h1.x);
  atomicAdd(op + 5, alpha * h1.y);
  atomicAdd(op + 6, alpha * h1.z);
  atomicAdd(op + 7, alpha * h1.w);
}

// Fused ReLU + LayerNorm(256), in place. One wave per row, 8 floats per lane.
__global__ __launch_bounds__(256) void relu_ln_kernel(
    float* __restrict__ out, const float* __restrict__ g,
    const float* __restrict__ b, int N)
{
  const int wave = threadIdx.x >> 5;
  const int lane = threadIdx.x & 31;
  const int row  = blockIdx.x * 8 + wave;
  if (row >= N) return;

  float* p = out + (size_t)row * 256 + lane * 8;
  float4 a = *(const float4*)(p);
  float4 c = *(const float4*)(p + 4);
  float v[8] = { a.x, a.y, a.z, a.w, c.x, c.y, c.z, c.w };
  float sum = 0.f, sumsq = 0.f;
  #pragma unroll
  for (int j = 0; j < 8; ++j) {
    v[j] = fmaxf(v[j], 0.0f);            // ReLU
    sum += v[j];
    sumsq += v[j] * v[j];
  }
  #pragma unroll
  for (int off = 16; off >= 1; off >>= 1) {
    sum   += __shfl_xor(sum, off, 32);
    sumsq += __shfl_xor(sumsq, off, 32);
  }
  float mu  = sum * (1.0f / 256.0f);
  float var = sumsq * (1.0f / 256.0f) - mu * mu;
  float inv = 1.0f / sqrtf(var + EPS_LN);
  const float* gp = g + lane * 8;
  const float* bp = b + lane * 8;
  float o[8];
  #pragma unroll
  for (int j = 0; j < 8; ++j) o[j] = (v[j] - mu) * inv * gp[j] + bp[j];
  float4 o0 = { o[0], o[1], o[2], o[3] };
  float4 o1 = { o[4], o[5], o[6], o[7] };
  *(float4*)(p)     = o0;
  *(float4*)(p + 4) = o1;
}

static inline int cdiv(long long a, long long b) { return (int)((a + b - 1) / b); }

extern "C" void kernel_launch(void* const* d_in, const int* in_sizes, int n_in,
                              void* d_out, int out_size, void* d_ws, size_t ws_size,
                              hipStream_t stream)
{
  const float* x_job     = (const float*)d_in[0];
  const float* x_station = (const float*)d_in[1];
  const float* x_machine = (const float*)d_in[2];
  const float* x_robot   = (const float*)d_in[3];
  const int NJ = in_sizes[0] / 128;
  const int NS = in_sizes[1] / 128;
  const int NM = in_sizes[2] / 128;
  const int NR = in_sizes[3] / 128;

  const float* xd_rel[5] = { x_station, x_station, x_machine, x_machine, x_robot };
  const int    nd_rel[5] = { NS, NS, NM, NM, NR };

  float* out_s = (float*)d_out;
  float* out_m = out_s + (size_t)NS * 256;
  float* out_r = out_m + (size_t)NM * 256;
  float* out_rel[5] = { out_s, out_s, out_m, out_m, out_r };

  int NDmax = NS > NM ? NS : NM;
  if (NR > NDmax) NDmax = NR;

  // Workspace layout (floats)
  float* ws   = (float*)d_ws;
  float* hs   = ws;                           // NJ*256  (reused per relation)
  float* al_s = hs + (size_t)NJ * 256;        // NJ*4
  float* al_d = al_s + (size_t)NJ * 4;        // NDmax*4
  float* mbuf = al_d + (size_t)NDmax * 4;     // NDmax*4
  float* zbuf = mbuf + (size_t)NDmax * 4;     // NDmax*4
  float* Wa_s = zbuf + (size_t)NDmax * 4;     // 512
  float* Wa_d = Wa_s + 512;                   // 512

  // Bias inits: out_s = b_cbl+b_li, out_m = b_nd+b_ex, out_r = b_hb
  const float* b_cbl = (const float*)d_in[8];
  const float* b_li  = (const float*)d_in[13];
  const float* b_nd  = (const float*)d_in[18];
  const float* b_ex  = (const float*)d_in[23];
  const float* b_hb  = (const float*)d_in[28];
  init_bias_kernel<<<cdiv((long long)NS * 256, 256), 256, 0, stream>>>(out_s, b_cbl, b_li, NS * 256);
  init_bias_kernel<<<cdiv((long long)NM * 256, 256), 256, 0, stream>>>(out_m, b_nd, b_ex, NM * 256);
  init_bias_kernel<<<cdiv((long long)NR * 256, 256), 256, 0, stream>>>(out_r, b_hb, nullptr, NR * 256);

  for (int r = 0; r < 5; ++r) {
    const int base = 4 + 5 * r;
    const int* ei  = (const int*)d_in[base];
    const int  E   = in_sizes[base] / 2;
    const int* src = ei;
    const int* dst = ei + E;
    const float* W     = (const float*)d_in[base + 1];
    const float* a_src = (const float*)d_in[base + 2];
    const float* a_dst = (const float*)d_in[base + 3];
    const int ND = nd_rel[r];

    dim3 gg(cdiv(NJ, 256), 4);
    gemm128x256_wmma<<<gg, 256, 0, stream>>>(x_job, W, hs, NJ, 0);
    compute_wa_kernel<<<1, 128, 0, stream>>>(W, a_src, a_dst, Wa_s, Wa_d);
    attn_logits_kernel<<<cdiv(NJ, 8), 256, 0, stream>>>(x_job, Wa_s, al_s, NJ);
    attn_logits_kernel<<<cdiv(ND, 8), 256, 0, stream>>>(xd_rel[r], Wa_d, al_d, ND);
    init_mz_kernel<<<cdiv(ND * 4, 256), 256, 0, stream>>>(mbuf, zbuf, ND * 4);
    edge_max_kernel<<<cdiv(E, 256), 256, 0, stream>>>(src, dst, al_s, al_d, mbuf, E);
    edge_expsum_kernel<<<cdiv(E, 256), 256, 0, stream>>>(src, dst, al_s, al_d, mbuf, zbuf, E);
    edge_msg_kernel<<<cdiv((long long)E * 32, 256), 256, 0, stream>>>(
        src, dst, al_s, al_d, mbuf, zbuf, hs, out_rel[r], E * 32);
  }

  // Residual GEMMs accumulate into d_out, then fused ReLU+LayerNorm in place.
  const float* Wres_s = (const float*)d_in[29];
  const float* g_s    = (const float*)d_in[30];
  const float* be_s   = (const float*)d_in[31];
  const float* Wres_m = (const float*)d_in[32];
  const float* g_m    = (const float*)d_in[33];
  const float* be_m   = (const float*)d_in[34];
  const float* Wres_r = (const float*)d_in[35];
  const float* g_r    = (const float*)d_in[36];
  const float* be_r   = (const float*)d_in[37];

  {
    dim3 gs(cdiv(NS, 256), 4), gm(cdiv(NM, 256), 4), gr(cdiv(NR, 256), 4);
    gemm128x256_wmma<<<gs, 256, 0, stream>>>(x_station, Wres_s, out_s, NS, 1);
    gemm128x256_wmma<<<gm, 256, 0, stream>>>(x_machine, Wres_m, out_m, NM, 1);
    gemm128x256_wmma<<<gr, 256, 0, stream>>>(x_robot,   Wres_r, out_r, NR, 1);
  }
  relu_ln_kernel<<<cdiv(NS, 8), 256, 0, stream>>>(out_s, g_s, be_s, NS);
  relu_ln_kernel<<<cdiv(NM, 8), 256, 0, stream>>>(out_m, g_m, be_m, NM);
  relu_ln_kernel<<<cdiv(NR, 8), 256, 0, stream>>>(out_r, g_r, be_r, NR);
}